// GAT_35012573397500
// MI455X (gfx1250) — compile-verified
//
#include <hip/hip_runtime.h>
#include <hip/hip_bf16.h>
#include <math.h>

typedef float v2f __attribute__((ext_vector_type(2)));
typedef float v8f __attribute__((ext_vector_type(8)));

#define NEG_SLOPE 0.2f

// ---------------------------------------------------------------------------
// GEMM: C[M,64] = A[M,K] @ B[K,64], via V_WMMA_F32_16X16X4_F32.
// One 16x16 output tile per wave; 4 waves per block cover the 64 columns.
// blockIdx.x = M-tile. EXEC is all-ones (no divergence before wmma).
// ---------------------------------------------------------------------------
__global__ __launch_bounds__(128)
void gat_gemm_wmma(const float* __restrict__ A, const float* __restrict__ B,
                   float* __restrict__ C, int Mrows, int K) {
    const int wave = threadIdx.x >> 5;      // 0..3 -> N tile (16 cols each)
    const int lane = threadIdx.x & 31;
    const int half = lane >> 4;             // 0: lanes 0-15, 1: lanes 16-31
    const int l16  = lane & 15;
    const int m0   = blockIdx.x * 16;
    const int n0   = wave * 16;

    // A fragment row (clamped for safety; M is an exact multiple of 16 here)
    int arow_i = m0 + l16;
    if (arow_i >= Mrows) arow_i = Mrows - 1;
    const float* __restrict__ arow = A + (size_t)arow_i * K;
    const float* __restrict__ bcol = B + n0 + l16;     // column n0+l16 of B

    v8f acc = {};
    for (int k = 0; k < K; k += 4) {
        const int ka = k + 2 * half;        // K index for VGPR0 of this lane
        v2f a;                              // A: lane holds (M, ka), (M, ka+1)
        a.x = arow[ka];
        a.y = arow[ka + 1];
        v2f b;                              // B: lane holds (ka, n), (ka+1, n)
        b.x = bcol[(size_t)ka * 64];
        b.y = bcol[(size_t)(ka + 1) * 64];
        acc = __builtin_amdgcn_wmma_f32_16x16x4_f32(
            false, a, false, b, (short)0, acc, false, false);
    }

    // D store: VGPR r -> row m0 + r + 8*half, col n0 + l16
    const int rbase = m0 + 8 * half;
#pragma unroll
    for (int r = 0; r < 8; ++r) {
        const int row = rbase + r;
        if (row < Mrows) C[(size_t)row * 64 + n0 + l16] = acc[r];
    }
}

// ---------------------------------------------------------------------------
// Per-node attention logits (one wave32 per node) + buffer init.
//   as[n] = h[n,:] . a_src ; ad[n] = h[n,:] . a_dst
//   m[n] = -inf ; denom[n] = 0 ; outacc[n,:] = 0
// ---------------------------------------------------------------------------
__global__ __launch_bounds__(256)
void gat_node_alpha_init(const float* __restrict__ h,
                         const float* __restrict__ a_src,
                         const float* __restrict__ a_dst,
                         float* __restrict__ as, float* __restrict__ ad,
                         float* __restrict__ m, float* __restrict__ denom,
                         float* __restrict__ outacc, int N) {
    const int node = blockIdx.x * (blockDim.x >> 5) + (threadIdx.x >> 5);
    const int lane = threadIdx.x & 31;
    if (node >= N) return;

    const float* __restrict__ hr = h + (size_t)node * 64;
    const float h0 = hr[lane];
    const float h1 = hr[lane + 32];
    float s = h0 * a_src[lane] + h1 * a_src[lane + 32];
    float d = h0 * a_dst[lane] + h1 * a_dst[lane + 32];
#pragma unroll
    for (int off = 16; off > 0; off >>= 1) {
        s += __shfl_down(s, off, 32);
        d += __shfl_down(d, off, 32);
    }
    outacc[(size_t)node * 64 + lane]      = 0.0f;
    outacc[(size_t)node * 64 + lane + 32] = 0.0f;
    if (lane == 0) {
        as[node]    = s;
        ad[node]    = d;
        m[node]     = -INFINITY;
        denom[node] = 0.0f;
    }
}

// Float atomic-max via signed/unsigned integer ordering trick (always lowers
// to global_atomic_max / global_atomic_min).
__device__ __forceinline__ void atomicMaxF(float* addr, float val) {
    if (val >= 0.0f)
        atomicMax((int*)addr, __float_as_int(val));
    else
        atomicMin((unsigned int*)addr, __float_as_uint(val));
}

__device__ __forceinline__ void edgeSrcDst(int e, int E,
                                           const int* __restrict__ src,
                                           const int* __restrict__ dst,
                                           int& s, int& d) {
    if (e < E) { s = src[e]; d = dst[e]; }
    else       { s = e - E;  d = e - E;  }   // self loops appended
}

// Pass 1: segment max of leaky_relu(as[src] + ad[dst]) over dst
__global__ __launch_bounds__(256)
void gat_edge_max(const int* __restrict__ src, const int* __restrict__ dst,
                  const float* __restrict__ as, const float* __restrict__ ad,
                  float* __restrict__ m, int E, int N) {
    const int e = blockIdx.x * blockDim.x + threadIdx.x;
    if (e >= E + N) return;
    int s, d;
    edgeSrcDst(e, E, src, dst, s, d);
    float sc = as[s] + ad[d];
    sc = (sc > 0.0f) ? sc : NEG_SLOPE * sc;
    atomicMaxF(&m[d], sc);
}

// Pass 2: ex = exp(score - m[dst]); denom[dst] += ex; stash ex per edge
__global__ __launch_bounds__(256)
void gat_edge_exp(const int* __restrict__ src, const int* __restrict__ dst,
                  const float* __restrict__ as, const float* __restrict__ ad,
                  const float* __restrict__ m, float* __restrict__ exw,
                  float* __restrict__ denom, int E, int N) {
    const int e = blockIdx.x * blockDim.x + threadIdx.x;
    if (e >= E + N) return;
    int s, d;
    edgeSrcDst(e, E, src, dst, s, d);
    float sc = as[s] + ad[d];
    sc = (sc > 0.0f) ? sc : NEG_SLOPE * sc;
    const float ex = __expf(sc - m[d]);
    exw[e] = ex;
    atomicAdd(&denom[d], ex);
}

// Pass 3: out[dst,:] += h[src,:] * (ex / denom[dst]); 64 lanes per edge
__global__ __launch_bounds__(256)
void gat_edge_scatter(const int* __restrict__ src, const int* __restrict__ dst,
                      const float* __restrict__ h, const float* __restrict__ exw,
                      const float* __restrict__ denom, float* __restrict__ out,
                      int E, int N) {
    const int t = blockIdx.x * blockDim.x + threadIdx.x;
    const int e = t >> 6;
    const int f = t & 63;
    if (e >= E + N) return;
    int s, d;
    edgeSrcDst(e, E, src, dst, s, d);
    const float alpha = exw[e] / denom[d];
    atomicAdd(&out[(size_t)d * 64 + f], h[(size_t)s * 64 + f] * alpha);
}

// Epilogue: out += bias; optional ELU
__global__ __launch_bounds__(256)
void gat_bias_act(float* __restrict__ out, const float* __restrict__ bias,
                  int total, int apply_elu) {
    const int i = blockIdx.x * blockDim.x + threadIdx.x;
    if (i >= total) return;
    float v = out[i] + bias[i & 63];
    if (apply_elu) v = (v > 0.0f) ? v : (__expf(v) - 1.0f);
    out[i] = v;
}

// ---------------------------------------------------------------------------
extern "C" void kernel_launch(void* const* d_in, const int* in_sizes, int n_in,
                              void* d_out, int out_size, void* d_ws, size_t ws_size,
                              hipStream_t stream) {
    const float* x   = (const float*)d_in[0];
    const int*   ei  = (const int*)  d_in[1];   // int32 (JAX default x64 off)
    const float* W1  = (const float*)d_in[2];
    const float* aS1 = (const float*)d_in[3];
    const float* aD1 = (const float*)d_in[4];
    const float* b1  = (const float*)d_in[5];
    const float* W2  = (const float*)d_in[6];
    const float* aS2 = (const float*)d_in[7];
    const float* aD2 = (const float*)d_in[8];
    const float* b2  = (const float*)d_in[9];

    const int H   = in_sizes[3];          // 64
    const int Fin = in_sizes[2] / H;      // 128
    const int N   = in_sizes[0] / Fin;    // 50000
    const int E   = in_sizes[1] / 2;      // 800000
    const int Et  = E + N;                // with self loops

    const int* srcIdx = ei;
    const int* dstIdx = ei + E;

    // workspace layout (floats)
    float* ws     = (float*)d_ws;
    float* h      = ws;                         // N*64  projected features
    float* outacc = h + (size_t)N * 64;         // N*64  layer-1 aggregate
    float* as     = outacc + (size_t)N * 64;    // N
    float* ad     = as + N;                     // N
    float* m      = ad + N;                     // N
    float* denom  = m + N;                      // N
    float* exw    = denom + N;                  // Et

    const int mtiles  = (N + 15) / 16;
    const int eBlocks = (Et + 255) / 256;
    const int sBlocks = (Et * 64 + 255) / 256;  // 54.4M threads, fits int
    const int nBlocks = ((size_t)N * 64 + 255) / 256;
    const int wBlocks = (N + 7) / 8;            // 8 waves per 256-thread block

    float* out = (float*)d_out;

    // -------- layer 1: h = x @ W1 ; aggregate -> outacc ; ELU --------
    gat_gemm_wmma<<<mtiles, 128, 0, stream>>>(x, W1, h, N, Fin);
    gat_node_alpha_init<<<wBlocks, 256, 0, stream>>>(h, aS1, aD1, as, ad, m,
                                                     denom, outacc, N);
    gat_edge_max    <<<eBlocks, 256, 0, stream>>>(srcIdx, dstIdx, as, ad, m, E, N);
    gat_edge_exp    <<<eBlocks, 256, 0, stream>>>(srcIdx, dstIdx, as, ad, m,
                                                  exw, denom, E, N);
    gat_edge_scatter<<<sBlocks, 256, 0, stream>>>(srcIdx, dstIdx, h, exw,
                                                  denom, outacc, E, N);
    gat_bias_act    <<<nBlocks, 256, 0, stream>>>(outacc, b1, N * 64, 1);

    // -------- layer 2: h = outacc @ W2 ; aggregate -> d_out ; +bias --------
    gat_gemm_wmma<<<mtiles, 128, 0, stream>>>(outacc, W2, h, N, H);
    gat_node_alpha_init<<<wBlocks, 256, 0, stream>>>(h, aS2, aD2, as, ad, m,
                                                     denom, out, N);
    gat_edge_max    <<<eBlocks, 256, 0, stream>>>(srcIdx, dstIdx, as, ad, m, E, N);
    gat_edge_exp    <<<eBlocks, 256, 0, stream>>>(srcIdx, dstIdx, as, ad, m,
                                                  exw, denom, E, N);
    gat_edge_scatter<<<sBlocks, 256, 0, stream>>>(srcIdx, dstIdx, h, exw,
                                                  denom, out, E, N);
    gat_bias_act    <<<nBlocks, 256, 0, stream>>>(out, b2, N * 64, 0);
}